// linear_gla_19189913879291
// MI455X (gfx1250) — compile-verified
//
#include <hip/hip_runtime.h>

#define B_  8
#define C_  512
#define G_  32
#define CG_ 16
#define H_  28
#define W_  28
#define HW_ 784
#define NTILE_ 49   // HW/16

typedef __attribute__((ext_vector_type(2))) float v2f;
typedef __attribute__((ext_vector_type(8))) float v8f;

// ---------- kernel 1: global average pool: y[b,c] = mean_hw x ----------
__global__ __launch_bounds__(128) void gla_gap_kernel(const float* __restrict__ x,
                                                      float* __restrict__ y) {
  __shared__ float red[128];
  int bc = blockIdx.x;                       // 0..4095
  const float* xp = x + (size_t)bc * HW_;
  float s = 0.f;
  for (int t = threadIdx.x; t < HW_; t += 128) s += xp[t];
  red[threadIdx.x] = s;
  __syncthreads();
  for (int off = 64; off > 0; off >>= 1) {
    if ((int)threadIdx.x < off) red[threadIdx.x] += red[threadIdx.x + off];
    __syncthreads();
  }
  if (threadIdx.x == 0) y[bc] = red[0] * (1.0f / HW_);
}

// ---------- kernel 2: Q/K 5-tap conv along c, elu+1, z_new, qz, QK ----------
__global__ __launch_bounds__(512) void gla_qk_kernel(const float* __restrict__ y,
                                                     const float* __restrict__ z,
                                                     const float* __restrict__ wq,
                                                     const float* __restrict__ wk,
                                                     float* __restrict__ Qp,
                                                     float* __restrict__ Kp,
                                                     float* __restrict__ qzp,
                                                     float* __restrict__ qkp,
                                                     float* __restrict__ znew_out) {
  __shared__ float sy[C_];
  __shared__ float sa[C_];
  __shared__ float sb[C_];
  int b = blockIdx.x;
  int c = threadIdx.x;
  sy[c] = y[b * C_ + c];
  __syncthreads();
  float q = 0.f, k = 0.f;
#pragma unroll
  for (int j = 0; j < 5; ++j) {
    int cc = c + j - 2;
    float yv = (cc >= 0 && cc < C_) ? sy[cc] : 0.f;
    q += yv * wq[j];
    k += yv * wk[j];
  }
  // elu(x)+1 : x>0 ? x+1 : exp(x)
  q = (q > 0.f) ? q + 1.f : expf(q);
  k = (k > 0.f) ? k + 1.f : expf(k);
  float zn = z[b * C_ + c] + k;
  znew_out[b * C_ + c] = zn;
  Qp[b * C_ + c] = q;
  Kp[b * C_ + c] = k;
  sa[c] = q * (zn + 1e-6f);
  sb[c] = q * k;
  __syncthreads();
  if ((c & 15) == 0) {
    float s1 = 0.f, s2 = 0.f;
#pragma unroll
    for (int i = 0; i < CG_; ++i) { s1 += sa[c + i]; s2 += sb[c + i]; }
    int g = c >> 4;
    qzp[b * G_ + g] = 1.0f / s1;
    qkp[b * G_ + g] = s2;
  }
}

// ---------- kernel 3: fused streaming pass over s (WMMA c-contraction) ----------
// wave handles one (b,g, 16-pixel tile):
//   s_new[c,s,d] = s_old + K[c]*V[s,d]       (V = inline depthwise 3x3 of x)
//   out[s,d]     = qz*( sum_c Q[c]*s_old[c,s,d] + QK*V[s,d] )
// WMMA f32 16x16x4: A = Q-chunk replicated over M rows, B = s_old tile (loads
// land directly in B layout: lane L -> n=L%16, reg r -> k = r + 2*(L/16)).
__global__ __launch_bounds__(256) void gla_main_kernel(
    const float* __restrict__ x, const float* __restrict__ sold,
    const float* __restrict__ wv,
    const float* __restrict__ Qp, const float* __restrict__ Kp,
    const float* __restrict__ qzp, const float* __restrict__ qkp,
    float* __restrict__ outp, float* __restrict__ snewp) {
  int lane = threadIdx.x & 31;
  int wid  = (int)((blockIdx.x * blockDim.x + threadIdx.x) >> 5);
  int tile = wid % NTILE_;
  int bg   = wid / NTILE_;                   // 0..255
  int g = bg & (G_ - 1);
  int b = bg >> 5;
  int hi = lane >> 4;                        // half-wave: k offset +2
  int dl = lane & 15;                        // pixel within tile (matrix N index)
  int d0 = tile * 16;
  int d  = d0 + dl;
  int pi = d / W_, pj = d % W_;

  // A operand (Q) and K values; lane L, reg r <-> c = kc*4 + 2*hi + r
  const float* Qb = Qp + b * C_ + g * CG_;
  const float* Kb = Kp + b * C_ + g * CG_;
  v2f   Av[4];
  float aK[4][2];
#pragma unroll
  for (int kc = 0; kc < 4; ++kc) {
    int c0 = kc * 4 + 2 * hi;
    Av[kc].x  = Qb[c0];
    Av[kc].y  = Qb[c0 + 1];
    aK[kc][0] = Kb[c0];
    aK[kc][1] = Kb[c0 + 1];
  }
  float qzv = qzp[bg];
  float qkv = qkp[bg];

  // 3x3 tap offsets / validity for this lane's pixel (independent of s)
  int  off9[9];
  bool ok9[9];
#pragma unroll
  for (int t = 0; t < 9; ++t) {
    int di = t / 3 - 1, dj = t % 3 - 1;
    int ii = pi + di, jj = pj + dj;
    bool ok = (ii >= 0 && ii < H_ && jj >= 0 && jj < W_);
    ok9[t]  = ok;
    off9[t] = ok ? (ii * W_ + jj) : 0;
  }

  const float* sbase  = sold  + (size_t)bg * (CG_ * CG_ * HW_) + d0 + dl;
  float*       snbase = snewp + (size_t)bg * (CG_ * CG_ * HW_) + d0 + dl;

#pragma unroll 1
  for (int s = 0; s < CG_; ++s) {
    int ch = g * CG_ + s;
    // V[s,d]: depthwise 3x3 conv (x stays hot in cache; DRAM-read once)
    const float* xb = x + (size_t)(b * C_ + ch) * HW_;
    const float* wb = wv + ch * 9;
    float v = 0.f;
#pragma unroll
    for (int t = 0; t < 9; ++t) {
      float xv = ok9[t] ? xb[off9[t]] : 0.f;
      v += xv * wb[t];
    }
    // stream in the 16(c) x 16(d) block of old state for this s
    float sv[4][2];
#pragma unroll
    for (int kc = 0; kc < 4; ++kc) {
      int c0 = kc * 4 + 2 * hi;
      sv[kc][0] = sbase[(c0    ) * (CG_ * HW_) + s * HW_];
      sv[kc][1] = sbase[(c0 + 1) * (CG_ * HW_) + s * HW_];
    }
    // acc[d] = sum_c Q[c]*s_old[c,s,d] via 4 chained f32 WMMAs (K=4 each)
    v8f acc = {0.f, 0.f, 0.f, 0.f, 0.f, 0.f, 0.f, 0.f};
#pragma unroll
    for (int kc = 0; kc < 4; ++kc) {
      v2f Bv;
      Bv.x = sv[kc][0];
      Bv.y = sv[kc][1];
      acc = __builtin_amdgcn_wmma_f32_16x16x4_f32(false, Av[kc], false, Bv,
                                                  (short)0, acc, false, false);
    }
    // rank-1 state update, write-once
#pragma unroll
    for (int kc = 0; kc < 4; ++kc) {
      int c0 = kc * 4 + 2 * hi;
      snbase[(c0    ) * (CG_ * HW_) + s * HW_] = sv[kc][0] + aK[kc][0] * v;
      snbase[(c0 + 1) * (CG_ * HW_) + s * HW_] = sv[kc][1] + aK[kc][1] * v;
    }
    // D row 0 lives in acc[0], lanes 0-15 (all rows identical since A rows are)
    float o = qzv * (acc[0] + qkv * v);
    if (lane < 16) outp[(size_t)(b * C_ + ch) * HW_ + d] = o;
  }
}

extern "C" void kernel_launch(void* const* d_in, const int* in_sizes, int n_in,
                              void* d_out, int out_size, void* d_ws, size_t ws_size,
                              hipStream_t stream) {
  (void)in_sizes; (void)n_in; (void)out_size; (void)ws_size;
  const float* x  = (const float*)d_in[0];
  const float* s  = (const float*)d_in[1];
  const float* z  = (const float*)d_in[2];
  const float* wq = (const float*)d_in[3];
  const float* wk = (const float*)d_in[4];
  const float* wv = (const float*)d_in[5];

  float* out  = (float*)d_out;                               // [8,512,28,28]
  float* snew = out + (size_t)B_ * C_ * HW_;                 // [8,32,16,16,784]
  float* znew = snew + (size_t)B_ * G_ * CG_ * CG_ * HW_;    // [8,1,32,16]

  float* ws = (float*)d_ws;     // ~51 KB of scratch
  float* y  = ws;               // 4096
  float* Qp = ws + 4096;        // 4096
  float* Kp = ws + 8192;        // 4096
  float* qz = ws + 12288;       // 256
  float* qk = ws + 12544;       // 256

  gla_gap_kernel<<<B_ * C_, 128, 0, stream>>>(x, y);
  gla_qk_kernel<<<B_, C_, 0, stream>>>(y, z, wq, wk, Qp, Kp, qz, qk, znew);
  int nwaves = B_ * G_ * NTILE_;               // 12544 waves, 8 per block
  gla_main_kernel<<<nwaves / 8, 256, 0, stream>>>(x, s, wv, Qp, Kp, qz, qk, out, snew);
}